// GCN_79946521248145
// MI455X (gfx1250) — compile-verified
//
#include <hip/hip_runtime.h>

typedef float v2f __attribute__((ext_vector_type(2)));
typedef float v8f __attribute__((ext_vector_type(8)));

// ---------------- CSR construction ----------------

__global__ void k_zero_i32(int* __restrict__ p, int n) {
    int i = blockIdx.x * blockDim.x + threadIdx.x;
    if (i < n) p[i] = 0;
}

__global__ void k_count(const int* __restrict__ dst, int* __restrict__ cnt, int E) {
    int e = blockIdx.x * blockDim.x + threadIdx.x;
    if (e < E) atomicAdd(&cnt[dst[e]], 1);
}

// Single-block exclusive scan over cnt[] -> rowstart[], cursor[]; also dinv = rsqrt(cnt+1)
__global__ void k_scan(const int* __restrict__ cnt, int* __restrict__ rowstart,
                       int* __restrict__ cursor, float* __restrict__ dinv, int n) {
    __shared__ int sm[1024];
    __shared__ int s_carry;
    const int tid = threadIdx.x;
    if (tid == 0) s_carry = 0;
    __syncthreads();
    for (int base = 0; base < n; base += 1024) {
        int i = base + tid;
        int v = (i < n) ? cnt[i] : 0;
        sm[tid] = v;
        __syncthreads();
        // Hillis-Steele inclusive scan
        for (int off = 1; off < 1024; off <<= 1) {
            int t = (tid >= off) ? sm[tid - off] : 0;
            __syncthreads();
            sm[tid] += t;
            __syncthreads();
        }
        int incl  = sm[tid];
        int carry = s_carry;
        __syncthreads();                 // all reads of s_carry done before update
        if (i < n) {
            int excl = carry + incl - v;
            rowstart[i] = excl;
            cursor[i]   = excl;
            dinv[i]     = rsqrtf((float)(v + 1));   // +1 self-loop; always > 0
        }
        if (tid == 1023) s_carry = carry + incl;
        __syncthreads();
    }
    if (tid == 0) rowstart[n] = s_carry;
}

__global__ void k_scatter(const int* __restrict__ src, const int* __restrict__ dst,
                          int* __restrict__ cursor, int* __restrict__ adj, int E) {
    int e = blockIdx.x * blockDim.x + threadIdx.x;
    if (e < E) {
        int d   = dst[e];
        int pos = atomicAdd(&cursor[d], 1);
        adj[pos] = src[e];
    }
}

// ---------------- fp32 WMMA GEMM: C[M x NOUT] = A[M x K] * W[K x NOUT] ----------------
// One wave computes a 16x16 tile of C using v_wmma_f32_16x16x4_f32, looping K in steps of 4.
// A layout (16x4 fp32): lanes 0-15 -> K={0,1}, lanes 16-31 -> K={2,3}, M = lane&15.
// B layout (4x16 fp32): VGPR v, lanes 0-15 -> K=v, lanes 16-31 -> K=v+2, N = lane&15.

template <int K, int NOUT>
__global__ void k_gemm_wmma(const float* __restrict__ A, const float* __restrict__ W,
                            float* __restrict__ C, int M) {
    const int lane = threadIdx.x & 31;
    const int wave = threadIdx.x >> 5;
    const int col0 = wave * 16;              // this wave's 16 output columns
    const int r16  = lane & 15;
    const int koff = (lane >> 4) * 2;        // 0 or 2
    const long m0  = (long)blockIdx.x * 16;

    long arow_idx = m0 + r16;
    if (arow_idx > M - 1) arow_idx = M - 1;  // clamp loads for tail tile (keeps EXEC all-1s)
    const float* arow = A + arow_idx * K;
    const float* wp   = W + (size_t)koff * NOUT + col0 + r16;

    v8f c = {};
#pragma unroll 4
    for (int k = 0; k < K; k += 4) {
        v2f a, b;
        a.x = arow[k + koff];
        a.y = arow[k + koff + 1];
        b.x = wp[(size_t)k * NOUT];
        b.y = wp[(size_t)(k + 1) * NOUT];
        c = __builtin_amdgcn_wmma_f32_16x16x4_f32(false, a, false, b, (short)0, c,
                                                  false, false);
    }
    // C layout: VGPR v -> M = v + 8*(lane>>4), N = lane&15
    const int crow0 = (lane >> 4) * 8;
    if (m0 + 16 <= (long)M) {
        // full tile: branch-free stores, constant immediate offsets
        float* cbase = C + (m0 + crow0) * NOUT + col0 + r16;
#pragma unroll
        for (int v = 0; v < 8; ++v) cbase[(size_t)v * NOUT] = c[v];
    } else {
#pragma unroll
        for (int v = 0; v < 8; ++v) {
            long mrow = m0 + crow0 + v;
            if (mrow < M) C[mrow * NOUT + col0 + r16] = c[v];
        }
    }
}

// ---------------- CSR aggregation (one wave per node) ----------------
// out[i] = act( dinv_i * ( sum_s dinv_s * h[s] + dinv_i * h[i] ) + bias )

__global__ void k_agg128_relu(const float* __restrict__ h, const int* __restrict__ rowstart,
                              const int* __restrict__ adj, const float* __restrict__ dinv,
                              const float* __restrict__ bias, float* __restrict__ out, int n) {
    const int lane = threadIdx.x & 31;
    const int node = blockIdx.x * (blockDim.x >> 5) + (threadIdx.x >> 5);
    if (node >= n) return;
    const float di = dinv[node];
    float4 a = ((const float4*)(h + (size_t)node * 128))[lane];
    float ax = di * a.x, ay = di * a.y, az = di * a.z, aw = di * a.w;
    const int beg = rowstart[node], end = rowstart[node + 1];
    int j0 = beg;
    // main chunks: full 32 edges, compile-time trip count for unrolling
    for (; j0 + 32 <= end; j0 += 32) {
        int   s_l = adj[j0 + lane];
        float d_l = dinv[s_l];
#pragma unroll 4
        for (int u = 0; u < 32; ++u) {
            int   s  = __shfl(s_l, u, 32);
            float ds = __shfl(d_l, u, 32);
            float4 v = ((const float4*)(h + (size_t)s * 128))[lane];
            ax += ds * v.x; ay += ds * v.y; az += ds * v.z; aw += ds * v.w;
        }
    }
    int rem = end - j0;
    if (rem > 0) {
        int   s_l = 0; float d_l = 0.0f;
        if (lane < rem) { s_l = adj[j0 + lane]; d_l = dinv[s_l]; }
        for (int u = 0; u < rem; ++u) {
            int   s  = __shfl(s_l, u, 32);
            float ds = __shfl(d_l, u, 32);
            float4 v = ((const float4*)(h + (size_t)s * 128))[lane];
            ax += ds * v.x; ay += ds * v.y; az += ds * v.z; aw += ds * v.w;
        }
    }
    float4 b = ((const float4*)bias)[lane];
    float4 r;
    r.x = fmaxf(di * ax + b.x, 0.0f);
    r.y = fmaxf(di * ay + b.y, 0.0f);
    r.z = fmaxf(di * az + b.z, 0.0f);
    r.w = fmaxf(di * aw + b.w, 0.0f);
    ((float4*)(out + (size_t)node * 128))[lane] = r;
}

__global__ void k_agg64(const float* __restrict__ h, const int* __restrict__ rowstart,
                        const int* __restrict__ adj, const float* __restrict__ dinv,
                        const float* __restrict__ bias, float* __restrict__ out, int n) {
    const int lane = threadIdx.x & 31;
    const int node = blockIdx.x * (blockDim.x >> 5) + (threadIdx.x >> 5);
    if (node >= n) return;
    const float di = dinv[node];
    float2 a = ((const float2*)(h + (size_t)node * 64))[lane];
    float ax = di * a.x, ay = di * a.y;
    const int beg = rowstart[node], end = rowstart[node + 1];
    int j0 = beg;
    for (; j0 + 32 <= end; j0 += 32) {
        int   s_l = adj[j0 + lane];
        float d_l = dinv[s_l];
#pragma unroll 4
        for (int u = 0; u < 32; ++u) {
            int   s  = __shfl(s_l, u, 32);
            float ds = __shfl(d_l, u, 32);
            float2 v = ((const float2*)(h + (size_t)s * 64))[lane];
            ax += ds * v.x; ay += ds * v.y;
        }
    }
    int rem = end - j0;
    if (rem > 0) {
        int   s_l = 0; float d_l = 0.0f;
        if (lane < rem) { s_l = adj[j0 + lane]; d_l = dinv[s_l]; }
        for (int u = 0; u < rem; ++u) {
            int   s  = __shfl(s_l, u, 32);
            float ds = __shfl(d_l, u, 32);
            float2 v = ((const float2*)(h + (size_t)s * 64))[lane];
            ax += ds * v.x; ay += ds * v.y;
        }
    }
    float2 b = ((const float2*)bias)[lane];
    float2 r;
    r.x = di * ax + b.x;
    r.y = di * ay + b.y;
    ((float2*)(out + (size_t)node * 64))[lane] = r;
}

// ---------------- host side ----------------

extern "C" void kernel_launch(void* const* d_in, const int* in_sizes, int n_in,
                              void* d_out, int out_size, void* d_ws, size_t ws_size,
                              hipStream_t stream) {
    const float* x  = (const float*)d_in[0];
    const int*   ei = (const int*)  d_in[1];
    const float* W1 = (const float*)d_in[2];
    const float* b1 = (const float*)d_in[3];
    const float* W2 = (const float*)d_in[4];
    const float* b2 = (const float*)d_in[5];

    const int DIN = 256, DH = 128, DOUT = 64;
    const int N = in_sizes[0] / DIN;
    const int E = in_sizes[1] / 2;
    const int* src = ei;
    const int* dst = ei + E;

    char*  ws  = (char*)d_ws;
    size_t off = 0;
    auto alloc = [&](size_t bytes) -> void* {
        void* p = ws + off;
        off += (bytes + 255) & ~(size_t)255;
        return p;
    };
    int*   cnt      = (int*)  alloc((size_t)N * 4);
    int*   rowstart = (int*)  alloc((size_t)(N + 1) * 4);
    int*   cursor   = (int*)  alloc((size_t)N * 4);
    float* dinv     = (float*)alloc((size_t)N * 4);
    int*   adj      = (int*)  alloc((size_t)E * 4);
    float* h1       = (float*)alloc((size_t)N * DH * 4);
    float* out1     = (float*)alloc((size_t)N * DH * 4);
    float* h2       = (float*)alloc((size_t)N * DOUT * 4);
    (void)ws_size; (void)n_in; (void)out_size;

    // 1) degrees + CSR by destination
    k_zero_i32<<<(N + 255) / 256, 256, 0, stream>>>(cnt, N);
    k_count  <<<(E + 255) / 256, 256, 0, stream>>>(dst, cnt, E);
    k_scan   <<<1, 1024, 0, stream>>>(cnt, rowstart, cursor, dinv, N);
    k_scatter<<<(E + 255) / 256, 256, 0, stream>>>(src, dst, cursor, adj, E);

    // 2) layer 1: dense transform (WMMA) + normalized aggregation + bias + relu
    k_gemm_wmma<256, 128><<<(N + 15) / 16, 8 * 32, 0, stream>>>(x, W1, h1, N);
    k_agg128_relu<<<(N + 7) / 8, 256, 0, stream>>>(h1, rowstart, adj, dinv, b1, out1, N);

    // 3) layer 2
    k_gemm_wmma<128, 64><<<(N + 15) / 16, 4 * 32, 0, stream>>>(out1, W2, h2, N);
    k_agg64<<<(N + 7) / 8, 256, 0, stream>>>(h2, rowstart, adj, dinv, b2, (float*)d_out, N);
}